// SNN_mfcc_31954556682914
// MI455X (gfx1250) — compile-verified
//
#include <hip/hip_runtime.h>

// ---------------------------------------------------------------------------
// SNN (LIF x2) on MI455X / gfx1250, wave32, fp32 WMMA (V_WMMA_F32_16X16X4_F32)
//
// T=345, B=128, IN=13, H=128, OUT=9
//
// Phase 1: cur1_all[t,b,h] = sum_c x[b,c,t]*W1[h,c] + b1[h]   (WMMA GEMM -> ws)
// Phase 2: serial scan over t; mem1/W2 register-resident in WMMA operand
//          layouts; fc2 = 32 fp32 WMMAs per step in 4 independent accumulator
//          chains; cur1 loads software-pipelined one step ahead.
// ---------------------------------------------------------------------------

typedef float v2f __attribute__((ext_vector_type(2)));
typedef float v8f __attribute__((ext_vector_type(8)));

#define T_STEPS 345
#define BATCH   128
#define IN_C    13
#define HID     128
#define OUTN    9

// ---------------------------------------------------------------------------
// Phase 1: fc1 GEMM. Rows = t*B+b (44160), cols = h (128), K = 13 (pad to 16).
// One wave per 16x16 tile; grid = (2760 row-tiles, 8 col-tiles).
// Padding handled with clamped unconditional loads x {0,1} masks (no EXEC
// branches around the WMMA chain). Two independent accumulator chains.
//
// A (16x4 fp32, 2 VGPRs): lanes 0-15 -> M=lane, K={0,1}; lanes 16-31 -> K={2,3}
// B (4x16 fp32, 2 VGPRs): lane%16 -> N, lane/16 selects K-pair {0,1}/{2,3}
// C (16x16 fp32, 8 VGPRs): lane holds N=lane%16, rows M = 8*(lane/16)+i
// ---------------------------------------------------------------------------
__global__ __launch_bounds__(32) void snn_fc1_wmma(
    const float* __restrict__ x,    // (B, IN, T)
    const float* __restrict__ W1,   // (H, IN)
    const float* __restrict__ b1,   // (H)
    float* __restrict__ cur1)       // (T*B, H) workspace
{
    const int lane = threadIdx.x & 31;
    const int m    = lane & 15;     // A row (batch) / B col (h offset)
    const int half = lane >> 4;     // selects K-pair

    const int R     = blockIdx.x * 16;          // row tile base; 8 tiles per t
    const int t     = R / BATCH;
    const int bbase = R % BATCH;
    const int hbase = blockIdx.y * 16;

    const int b = bbase + m;
    const int h = hbase + m;
    const float* xrow  = x  + (size_t)b * (IN_C * T_STEPS) + t;
    const float* w1row = W1 + (size_t)h * IN_C;

    v8f acc0 = {}, acc1 = {};
#pragma unroll
    for (int kc = 0; kc < 4; ++kc) {
        const int c0 = 4 * kc + 2 * half;
        const int c1 = c0 + 1;
        // clamp index, zero via mask (values are finite -> mask*val is exact 0)
        const int   c0c = (c0 < IN_C) ? c0 : (IN_C - 1);
        const int   c1c = (c1 < IN_C) ? c1 : (IN_C - 1);
        const float m0  = (c0 < IN_C) ? 1.0f : 0.0f;
        const float m1_ = (c1 < IN_C) ? 1.0f : 0.0f;
        v2f a, w;
        a.x = m0  * xrow[c0c * T_STEPS];
        a.y = m1_ * xrow[c1c * T_STEPS];
        w.x = m0  * w1row[c0c];
        w.y = m1_ * w1row[c1c];
        if (kc & 1)
            acc1 = __builtin_amdgcn_wmma_f32_16x16x4_f32(
                false, a, false, w, (short)0, acc1, false, false);
        else
            acc0 = __builtin_amdgcn_wmma_f32_16x16x4_f32(
                false, a, false, w, (short)0, acc0, false, false);
    }
    const v8f acc = acc0 + acc1;

    const float bias = b1[h];
#pragma unroll
    for (int i = 0; i < 8; ++i) {
        const int row = R + 8 * half + i;           // = t*128 + batch
        cur1[row * HID + hbase + m] = acc[i] + bias;
    }
}

// ---------------------------------------------------------------------------
// Phase 2: sequential LIF scan. 8 blocks x 1 wave; block owns 16 batches.
// mem1 register-resident in WMMA A-operand layout:
//   lane (m,half) holds batch bbase+m, h = 4*kc + 2*half + {0,1}, kc=0..31
// W2 register-resident in B-operand layout (n=lane%16 -> out channel).
// Per step: fused LIF1 + 32 WMMAs (4 independent chains) with step-(t+1)
// cur1 loads issued chunk-by-chunk under the WMMA chain.
// ---------------------------------------------------------------------------
__global__ __launch_bounds__(32) void snn_scan_wmma(
    const float* __restrict__ cur1, // (T*B, H)
    const float* __restrict__ W2,   // (OUT, H)
    const float* __restrict__ b2,   // (OUT)
    const float* __restrict__ pbeta1, const float* __restrict__ pthr1,
    const float* __restrict__ pbeta2, const float* __restrict__ pthr2,
    float* __restrict__ out)        // spk_rec (T,B,OUT) ++ mem_rec (T,B,OUT)
{
    const int lane = threadIdx.x & 31;
    const int n    = lane & 15;     // batch offset (A rows) & out channel (B cols)
    const int half = lane >> 4;
    const int bbase = blockIdx.x * 16;

    const float bt1  = fminf(fmaxf(pbeta1[0], 0.0f), 1.0f);
    const float thr1 = pthr1[0];
    const float bt2  = fminf(fmaxf(pbeta2[0], 0.0f), 1.0f);
    const float thr2 = pthr2[0];

    // W2 in B-operand layout (clamped row + mask: no EXEC-divergent loads)
    const int   nc    = (n < OUTN) ? n : (OUTN - 1);
    const float nmask = (n < OUTN) ? 1.0f : 0.0f;
    const float* w2row = W2 + (size_t)nc * HID + 2 * half;
    v2f w2op[32];
#pragma unroll
    for (int kc = 0; kc < 32; ++kc) {
        v2f w = *(const v2f*)(w2row + 4 * kc);
        w2op[kc].x = nmask * w.x;
        w2op[kc].y = nmask * w.y;
    }
    const float b2v = nmask * b2[nc];

    v2f m1[32];                     // mem1 in A-operand layout
#pragma unroll
    for (int kc = 0; kc < 32; ++kc) { m1[kc].x = 0.0f; m1[kc].y = 0.0f; }
    float m2[8];                    // mem2 in C layout (batch bbase+8*half+i, o=n)
#pragma unroll
    for (int i = 0; i < 8; ++i) m2[i] = 0.0f;

    // this lane's cur1 row: batch bbase+n, h-offset 2*half (+4*kc per chunk)
    const float* rowbase = cur1 + (size_t)(bbase + n) * HID + 2 * half;
    const int TBO = T_STEPS * BATCH * OUTN;

    // software pipeline: preload step 0 currents into registers
    v2f cbuf[32];
#pragma unroll
    for (int kc = 0; kc < 32; ++kc)
        cbuf[kc] = *(const v2f*)(rowbase + 4 * kc);

    for (int t = 0; t < T_STEPS; ++t) {
        // next step's row (clamped on the last step; redundant load, in-bounds)
        const int tn = (t + 1 < T_STEPS) ? (t + 1) : t;
        const float* nrow = rowbase + (size_t)tn * BATCH * HID;

        v8f acc0 = {}, acc1 = {}, acc2 = {}, acc3 = {};
#pragma unroll
        for (int kc = 0; kc < 32; ++kc) {
            const v2f c = cbuf[kc];
            cbuf[kc] = *(const v2f*)(nrow + 4 * kc);   // prefetch t+1 under WMMAs

            // LIF-1: integrate, fire, subtract-reset
            const float mx = fmaf(bt1, m1[kc].x, c.x);
            const float my = fmaf(bt1, m1[kc].y, c.y);
            const float sx = (mx > thr1) ? 1.0f : 0.0f;
            const float sy = (my > thr1) ? 1.0f : 0.0f;
            m1[kc].x = fmaf(-sx, thr1, mx);
            m1[kc].y = fmaf(-sy, thr1, my);

            v2f a; a.x = sx; a.y = sy;
            // 4 independent accumulation chains on the matrix pipe
            if      ((kc & 3) == 0)
                acc0 = __builtin_amdgcn_wmma_f32_16x16x4_f32(
                    false, a, false, w2op[kc], (short)0, acc0, false, false);
            else if ((kc & 3) == 1)
                acc1 = __builtin_amdgcn_wmma_f32_16x16x4_f32(
                    false, a, false, w2op[kc], (short)0, acc1, false, false);
            else if ((kc & 3) == 2)
                acc2 = __builtin_amdgcn_wmma_f32_16x16x4_f32(
                    false, a, false, w2op[kc], (short)0, acc2, false, false);
            else
                acc3 = __builtin_amdgcn_wmma_f32_16x16x4_f32(
                    false, a, false, w2op[kc], (short)0, acc3, false, false);
        }
        const v8f acc = (acc0 + acc1) + (acc2 + acc3);

        // LIF-2 + write spk_rec / mem_rec (only out channels 0..8 are real)
        float spk2[8];
#pragma unroll
        for (int i = 0; i < 8; ++i) {
            const float cu = acc[i] + b2v;
            float mm = fmaf(bt2, m2[i], cu);
            const float s = (mm > thr2) ? 1.0f : 0.0f;
            mm = fmaf(-s, thr2, mm);
            m2[i] = mm;
            spk2[i] = s;
        }
        if (n < OUTN) {
            const int base = (t * BATCH + bbase + 8 * half) * OUTN + n;
#pragma unroll
            for (int i = 0; i < 8; ++i) {
                out[base + i * OUTN]       = spk2[i];
                out[TBO + base + i * OUTN] = m2[i];
            }
        }
    }
}

// ---------------------------------------------------------------------------
extern "C" void kernel_launch(void* const* d_in, const int* in_sizes, int n_in,
                              void* d_out, int out_size, void* d_ws, size_t ws_size,
                              hipStream_t stream) {
    (void)in_sizes; (void)n_in; (void)out_size; (void)ws_size;
    const float* x     = (const float*)d_in[0];
    const float* W1    = (const float*)d_in[1];
    const float* b1    = (const float*)d_in[2];
    const float* W2    = (const float*)d_in[3];
    const float* b2    = (const float*)d_in[4];
    const float* beta1 = (const float*)d_in[5];
    const float* thr1  = (const float*)d_in[6];
    const float* beta2 = (const float*)d_in[7];
    const float* thr2  = (const float*)d_in[8];

    float* cur1 = (float*)d_ws;     // needs T*B*H*4 = 22.6 MB

    // Phase 1: 2760 row-tiles x 8 h-tiles, one wave each
    dim3 g1((T_STEPS * BATCH) / 16, HID / 16);
    snn_fc1_wmma<<<g1, 32, 0, stream>>>(x, W1, b1, cur1);

    // Phase 2: 8 persistent waves, 16 batches each
    snn_scan_wmma<<<BATCH / 16, 32, 0, stream>>>(cur1, W2, b2,
                                                 beta1, thr1, beta2, thr2,
                                                 (float*)d_out);
}